// PillarFeatureNet_31550829757169
// MI455X (gfx1250) — compile-verified
//
#include <hip/hip_runtime.h>

// PillarFeatureNet forward for MI455X (gfx1250, wave32).
// Memory-bound (77 MB traffic vs 4.1 GFLOP @ 23.3 TB/s -> ~3.3us floor):
//  - f32 WMMA (V_WMMA_F32_16X16X4_F32) for the exact-precision 10x64 GEMM
//  - TDM (tensor_load_to_lds + s_wait_tensorcnt) DMAs the 8KB feature tile
//    straight into LDS, bypassing the VGPR file; TDM OOB-zero semantics
//    implement the tail zero-fill for free
//  - GELU quasi-convexity: max_m gelu(x_m) == max(gelu(min x), gelu(max x)),
//    so only 2 erf evaluations per output element (exact, not approximate)

#define MPTS    32
#define CRAW    4
#define CIN     10
#define CPAD    12            // K padded to multiple of 4
#define COUT    64
#define PPB     16            // pillars per block
#define ROWS    (PPB * MPTS)  // 512 GEMM rows per block
#define THREADS 256           // 8 waves
#define NWAVES  (THREADS / 32)

typedef __attribute__((ext_vector_type(2))) float v2f;
typedef __attribute__((ext_vector_type(4))) float v4f;
typedef __attribute__((ext_vector_type(8))) float v8f;
typedef __attribute__((ext_vector_type(4))) unsigned int u32x4;
typedef __attribute__((ext_vector_type(4))) int i32x4;
typedef __attribute__((ext_vector_type(8))) int i32x8;

#if __has_builtin(__builtin_amdgcn_tensor_load_to_lds) && \
    __has_builtin(__builtin_amdgcn_s_wait_tensorcnt)
#define HAVE_TDM 1
#else
#define HAVE_TDM 0
#endif

__device__ __forceinline__ float gelu_exact(float x) {
    return 0.5f * x * (1.0f + erff(x * 0.70710678118654752f));
}

__global__ __launch_bounds__(THREADS) void pfn_wmma_kernel(
    const float* __restrict__ features,   // (N, 32, 4)
    const int*   __restrict__ num_points, // (N,)
    const int*   __restrict__ coors,      // (N, 4) [z,y,x,?]
    const float* __restrict__ W,          // (64, 10)
    const float* __restrict__ gamma,
    const float* __restrict__ beta,
    const float* __restrict__ rmean,
    const float* __restrict__ rvar,
    float*       __restrict__ out,        // (N, 64)
    int nPillars)
{
    __shared__ float raw[ROWS][CRAW];        // 8 KB, TDM destination
    __shared__ float feats[ROWS][CPAD];      // 24 KB, augmented+masked A panel
    __shared__ float Bt[CPAD][COUT + 1];     // W^T, K-padded, +1 anti-conflict pad
    __shared__ float sums[PPB][3];
    __shared__ float bnScale[COUT];
    __shared__ float bnBias[COUT];
    __shared__ float invNp[PPB];
    __shared__ float cOff[PPB][3];
    __shared__ int   npLds[PPB];

    const int tid  = threadIdx.x;
    const int lane = tid & 31;
    const int wave = tid >> 5;
    const int pillarBase = blockIdx.x * PPB;
    const int validRows  = min(ROWS, (nPillars - pillarBase) * MPTS);

    // ---- Phase 0a: DMA the raw feature tile into LDS via the Tensor Data
    // Mover (one instruction, wave 0 only; TENSORcnt tracks completion).
#if HAVE_TDM
    if (wave == 0) {
        const unsigned long long gaddr =
            (unsigned long long)(uintptr_t)features +
            (unsigned long long)pillarBase * (MPTS * CRAW * 4ull);
        const unsigned int laddr = (unsigned int)(uintptr_t)&raw[0][0];
        const unsigned int elems = (unsigned int)validRows * CRAW; // valid f32s
        const unsigned int tileE = ROWS * CRAW;                    // 2048 f32s

        u32x4 g0 = {};
        g0[0] = 1u;                                         // count=1, user mode
        g0[1] = laddr;                                      // lds_addr
        g0[2] = (unsigned int)(gaddr & 0xffffffffu);        // global_addr lo
        g0[3] = (unsigned int)((gaddr >> 32) & 0x01ffffffu) // global_addr[56:32]
              | (2u << 30);                                 // type=2 ("image")

        i32x8 g1 = {};
        g1[0] = (int)(2u << 16);                            // data_size=4B; mask=0
        g1[1] = (int)((elems & 0xffffu) << 16);             // tensor_dim0 lo16
        g1[2] = (int)((elems >> 16) & 0xffffu)              // tensor_dim0 hi16
              | (1 << 16);                                  // tensor_dim1 = 1
        g1[3] = (int)((tileE & 0xffffu) << 16);             // tile_dim0 = 2048
        g1[4] = 1;                                          // tile_dim1=1, tile_dim2=0
        g1[5] = (int)tileE;                                 // tensor_dim0_stride lo
        g1[6] = 0;                                          // stride hi, dim1_stride lo
        g1[7] = 0;

        i32x4 g2 = {};
        i32x4 g3 = {};
#if defined(__clang_major__) && (__clang_major__ >= 23)
        i32x8 g4 = {};
        __builtin_amdgcn_tensor_load_to_lds(g0, g1, g2, g3, g4, 0);
#else
        __builtin_amdgcn_tensor_load_to_lds(g0, g1, g2, g3, 0);
#endif
        __builtin_amdgcn_s_wait_tensorcnt(0);
    }
#else
    {
        const v4f* __restrict__ src = (const v4f*)features + (size_t)pillarBase * MPTS;
        v4f* dst = (v4f*)&raw[0][0];
        for (int i = tid; i < ROWS; i += THREADS) {
            v4f z = {0.0f, 0.0f, 0.0f, 0.0f};
            dst[i] = (i < validRows) ? src[i] : z;
        }
    }
#endif

    // ---- Phase 0b (overlaps the DMA): W^T panel, BN constants, pillar meta
    for (int i = tid; i < CPAD * COUT; i += THREADS) {
        int k = i / COUT, o = i % COUT;
        Bt[k][o] = (k < CIN) ? W[o * CIN + k] : 0.0f;
    }
    if (tid < COUT) {
        float s = rsqrtf(rvar[tid] + 1.0e-3f) * gamma[tid];
        bnScale[tid] = s;
        bnBias[tid]  = beta[tid] - rmean[tid] * s;
    }
    if (tid < PPB) {
        int g = pillarBase + tid;
        if (g < nPillars) {
            int np = num_points[g];
            npLds[tid] = np;
            invNp[tid] = 1.0f / (float)np;
            float cz = (float)coors[g * 4 + 0];
            float cy = (float)coors[g * 4 + 1];
            float cx = (float)coors[g * 4 + 2];
            cOff[tid][0] = cx * 0.2f + 0.1f;    // VX/2 + 0.0
            cOff[tid][1] = cy * 0.2f - 39.9f;   // VY/2 - 40.0
            cOff[tid][2] = cz * 4.0f - 1.0f;    // VZ/2 - 3.0
        } else {
            npLds[tid] = 0; invNp[tid] = 0.0f;
            cOff[tid][0] = cOff[tid][1] = cOff[tid][2] = 0.0f;
        }
    }
    __syncthreads();

    // ---- Phase 1: per-pillar xyz sums over ALL 32 rows (wave32 shfl reduction)
    for (int p = wave; p < PPB; p += NWAVES) {
        int r = p * MPTS + lane;           // lane == point index m
        float sx = raw[r][0], sy = raw[r][1], sz = raw[r][2];
        #pragma unroll
        for (int d = 16; d >= 1; d >>= 1) {
            sx += __shfl_xor(sx, d, 32);
            sy += __shfl_xor(sy, d, 32);
            sz += __shfl_xor(sz, d, 32);
        }
        if (lane == 0) { sums[p][0] = sx; sums[p][1] = sy; sums[p][2] = sz; }
    }
    __syncthreads();

    // ---- Phase 2: build augmented, masked, K-padded A panel in LDS
    for (int r = tid; r < ROWS; r += THREADS) {
        int p = r >> 5, m = r & 31;
        float f0 = raw[r][0], f1 = raw[r][1], f2 = raw[r][2], f3 = raw[r][3];
        float inp = invNp[p];
        float msk = (m < npLds[p]) ? 1.0f : 0.0f;
        feats[r][0] = f0 * msk;
        feats[r][1] = f1 * msk;
        feats[r][2] = f2 * msk;
        feats[r][3] = f3 * msk;
        feats[r][4] = (f0 - sums[p][0] * inp) * msk;
        feats[r][5] = (f1 - sums[p][1] * inp) * msk;
        feats[r][6] = (f2 - sums[p][2] * inp) * msk;
        feats[r][7] = (f0 - cOff[p][0]) * msk;
        feats[r][8] = (f1 - cOff[p][1]) * msk;
        feats[r][9] = (f2 - cOff[p][2]) * msk;
        feats[r][10] = 0.0f;
        feats[r][11] = 0.0f;
    }
    __syncthreads();

    // ---- Phase 3: f32 WMMA GEMM + BN + exact GELU + max over M
    // A 16x4 layout (ISA 7.12.2): lanes 0-15 -> K=kb,kb+1 ; lanes 16-31 ->
    // K=kb+2,kb+3 ; M = lane%16.  B mirrors with N = lane%16.
    const int nsel   = lane & 15;
    const int hiHalf = lane >> 4;
    const int kOfs   = hiHalf * 2;

    // B fragments are invariant across pillars: load once per wave.
    v2f b[4][3];
    #pragma unroll
    for (int ct = 0; ct < 4; ++ct) {
        const int o = ct * 16 + nsel;
        #pragma unroll
        for (int ks = 0; ks < 3; ++ks) {
            const int k = ks * 4 + kOfs;
            b[ct][ks].x = Bt[k][o];
            b[ct][ks].y = Bt[k + 1][o];
        }
    }

    for (int pi = 0; pi < 2; ++pi) {
        const int p = wave * 2 + pi;       // 8 waves x 2 = 16 pillars
        const int rowBase = p * MPTS;

        // A fragments: 2 half-pillar row tiles x 3 K-steps (reused over 4 col tiles)
        v2f a[2][3];
        #pragma unroll
        for (int h = 0; h < 2; ++h) {
            const int row = rowBase + h * 16 + nsel;
            #pragma unroll
            for (int ks = 0; ks < 3; ++ks) {
                const int k = ks * 4 + kOfs;
                a[h][ks].x = feats[row][k];
                a[h][ks].y = feats[row][k + 1];
            }
        }

        #pragma unroll
        for (int ct = 0; ct < 4; ++ct) {
            const int o = ct * 16 + nsel;
            v8f c0 = {}; v8f c1 = {};
            #pragma unroll
            for (int ks = 0; ks < 3; ++ks) {
                c0 = __builtin_amdgcn_wmma_f32_16x16x4_f32(
                        false, a[0][ks], false, b[ct][ks], (short)0, c0, false, false);
                c1 = __builtin_amdgcn_wmma_f32_16x16x4_f32(
                        false, a[1][ks], false, b[ct][ks], (short)0, c1, false, false);
            }

            // Min/max of the 16 raw accumulator values held by this lane.
            float cmin = fminf(c0[0], c1[0]);
            float cmax = fmaxf(c0[0], c1[0]);
            #pragma unroll
            for (int j = 1; j < 8; ++j) {
                cmin = fminf(cmin, fminf(c0[j], c1[j]));
                cmax = fmaxf(cmax, fmaxf(c0[j], c1[j]));
            }
            // lanes l and l+16 hold rows M and M+8 of each tile -> merge
            cmin = fminf(cmin, __shfl_xor(cmin, 16, 32));
            cmax = fmaxf(cmax, __shfl_xor(cmax, 16, 32));

            // BN affine on the two extremes only (sort handles scale sign),
            // then GELU at the endpoints: gelu is decreasing then increasing
            // (single minimum at x ~ -0.752), so the max over the 32 rows is
            // attained at one of the two extreme pre-activation values.
            const float s  = bnScale[o];
            const float bb = bnBias[o];
            const float x1 = fmaf(cmin, s, bb);
            const float x2 = fmaf(cmax, s, bb);
            const float lo = fminf(x1, x2);
            const float hi = fmaxf(x1, x2);
            const float r  = fmaxf(gelu_exact(lo), gelu_exact(hi));

            const int gp = pillarBase + p;
            if (lane < 16 && gp < nPillars)
                out[(size_t)gp * COUT + o] = r;
        }
    }
}

extern "C" void kernel_launch(void* const* d_in, const int* in_sizes, int n_in,
                              void* d_out, int out_size, void* d_ws, size_t ws_size,
                              hipStream_t stream) {
    const float* features   = (const float*)d_in[0];
    const int*   num_points = (const int*)  d_in[1];
    const int*   coors      = (const int*)  d_in[2];
    const float* W          = (const float*)d_in[3];
    const float* gamma      = (const float*)d_in[4];
    const float* beta       = (const float*)d_in[5];
    const float* rmean      = (const float*)d_in[6];
    const float* rvar       = (const float*)d_in[7];
    float*       out        = (float*)d_out;

    const int nPillars = in_sizes[1];                 // N from num_points
    const int blocks   = (nPillars + PPB - 1) / PPB;  // 6250 for N=100000

    pfn_wmma_kernel<<<blocks, THREADS, 0, stream>>>(
        features, num_points, coors, W, gamma, beta, rmean, rvar, out, nPillars);
}